// Router_23510650978886
// MI455X (gfx1250) — compile-verified
//
#include <hip/hip_runtime.h>

// ---------------------------------------------------------------------------
// Router: y = Conv1d(1, 8, k=32, stride=32)(x).sum(positions)
// == rowwise patch-column sums [B,32] then [32 -> 8] projection.
//
// Roofline: 512 MB fp32 streamed once from HBM @ 23.3 TB/s => ~22 us floor;
// math is negligible. Strategy: one fused streaming kernel, NT b128 loads,
// prefetch ahead, wave-level shuffle reduction, and the final 32->8 matmul
// done with V_WMMA_F32_16X16X4_F32 (M=16 rows per block tile, N padded to 16,
// K=32 in 8 steps of 4). B fragments are branch-free (clamped address +
// multiply mask) and preloaded before the streaming loop so the WMMA tail
// issues back-to-back with no EXEC toggling / xcnt stalls.
// ---------------------------------------------------------------------------

typedef __attribute__((ext_vector_type(2))) float v2f;
typedef __attribute__((ext_vector_type(4))) float v4f;
typedef __attribute__((ext_vector_type(8))) float v8f;

#define D_DIM  8192
#define KK     32          // conv kernel == patch width
#define OUTC   8
#define ROWS   16          // rows per block == waves per block == WMMA M
#define ITERS  (D_DIM / (32 * 4))   // 64 float4 iters per row per wave

__global__ __launch_bounds__(512)
void router_fused_kernel(const float* __restrict__ x,
                         const float* __restrict__ W,   // [8][32] flat
                         float* __restrict__ out,       // [B][8]
                         int B) {
    __shared__ __align__(16) float lds[ROWS * KK];      // 16 rows x 32 colsums

    const int lane = threadIdx.x & 31;
    const int wave = threadIdx.x >> 5;
    const int b0   = blockIdx.x * ROWS;
    const int row  = b0 + wave;

    const int n  = lane & 15;    // WMMA tile column (output channel; n<8 valid)
    const int hi = lane >> 4;    // lane half selects K sub-pair per ISA layout

    // -------- Preload B fragments (branch-free; same W for every block) -----
    // B 4x16 f32 layout: column n = lane%16; lanes 0-15 hold K {k0,k0+1},
    // lanes 16-31 hold K {k0+2,k0+3}. Pad N=8..15 via multiply mask so the
    // loads are unconditional b64s (address clamped with n&7, always valid).
    const float wmask = (n < OUTC) ? 1.0f : 0.0f;
    const v2f* __restrict__ W2 = (const v2f*)W;
    v2f bfrag[8];
    #pragma unroll
    for (int kc = 0; kc < 8; ++kc) {
        v2f w = W2[((lane & 7) * KK + kc * 4 + 2 * hi) >> 1];
        bfrag[kc].x = w.x * wmask;
        bfrag[kc].y = w.y * wmask;
    }

    // ---------------- Phase 1: streaming patch-column reduction -------------
    // Element offset for (it, lane, comp c) = it*128 + lane*4 + c.
    // Patch column = (lane*4 + c) % 32 = (lane%8)*4 + c  (iteration-invariant).
    const v4f* __restrict__ xr = (const v4f*)(x + (size_t)row * D_DIM);
    v4f acc = (v4f)0.0f;
    for (int it = 0; it < ITERS; it += 8) {
        if (it + 16 < ITERS)
            __builtin_prefetch(&xr[(it + 16) * 32 + lane], 0, 0);  // global_prefetch_b8
        #pragma unroll
        for (int u = 0; u < 8; ++u) {
            v4f v = __builtin_nontemporal_load(&xr[(it + u) * 32 + lane]);
            acc += v;
        }
    }
    // Lanes {l, l+8, l+16, l+24} hold partials for the same 4 columns:
    // reduce with xor-shuffles over masks 8 and 16 (wave32).
    #pragma unroll
    for (int m = 8; m <= 16; m <<= 1) {
        acc.x += __shfl_xor(acc.x, m, 32);
        acc.y += __shfl_xor(acc.y, m, 32);
        acc.z += __shfl_xor(acc.z, m, 32);
        acc.w += __shfl_xor(acc.w, m, 32);
    }
    if (lane < 8)
        ((v4f*)lds)[wave * 8 + lane] = acc;   // lds[wave][lane*4 .. +3]
    __syncthreads();

    // ---------------- Phase 2: [16x32] x [32x16(pad)] via WMMA --------------
    // Wave 0 only; branch is wave-uniform so EXEC is all ones inside (WMMA req).
    if (threadIdx.x < 32) {
        // Gather all A fragments first, then issue the WMMA chain back-to-back.
        // A 16x4 f32: lane L holds row M=L%16; VGPR0/1 = K {k0,k0+1}
        // (lanes 0-15) or K {k0+2,k0+3} (lanes 16-31).
        const v2f* __restrict__ lds2 = (const v2f*)lds;
        v2f afrag[8];
        #pragma unroll
        for (int kc = 0; kc < 8; ++kc)
            afrag[kc] = lds2[(n * KK + kc * 4 + 2 * hi) >> 1];   // ds_load_b64

        v8f c = (v8f)0.0f;
        #pragma unroll
        for (int kc = 0; kc < 8; ++kc) {
            c = __builtin_amdgcn_wmma_f32_16x16x4_f32(
                    /*neg_a=*/false, afrag[kc], /*neg_b=*/false, bfrag[kc],
                    /*c_mod=*/(short)0, c, /*reuse_a=*/false, /*reuse_b=*/false);
        }

        // D 16x16 f32: VGPR r -> row (r + 8*hi), column n = lane%16.
        if (n < OUTC) {
            #pragma unroll
            for (int r = 0; r < 8; ++r) {
                out[(size_t)(b0 + r + 8 * hi) * OUTC + n] = c[r];
            }
        }
    }
}

extern "C" void kernel_launch(void* const* d_in, const int* in_sizes, int n_in,
                              void* d_out, int out_size, void* d_ws, size_t ws_size,
                              hipStream_t stream) {
    const float* x = (const float*)d_in[0];   // [B, 8192] fp32
    const float* W = (const float*)d_in[1];   // [8, 1, 32] fp32
    float* out     = (float*)d_out;           // [B, 8] fp32

    const int B = in_sizes[0] / D_DIM;        // 16384
    dim3 grid(B / ROWS);                      // 1024 blocks of 512 threads
    router_fused_kernel<<<grid, 512, 0, stream>>>(x, W, out, B);
}